// ALIF_81166291959850
// MI455X (gfx1250) — compile-verified
//
#include <hip/hip_runtime.h>
#include <cstdint>

// ALIF spiking-neuron scan, MI455X (gfx1250).
// Bandwidth-bound: 64 MB in + 64 MB out -> ~5.5 us floor @ 23.3 TB/s.
// CDNA5 path: async global->LDS data mover (ASYNCcnt FIFO), 16 timesteps
// in flight per wave with zero VGPR cost; 4-step groups amortize waits.

constexpr int BLOCK  = 256;  // threads per block (8 wave32s)
constexpr int STAGES = 16;   // async FIFO depth (power of two, multiple of 4)

__global__ __launch_bounds__(BLOCK)
void alif_scan_kernel(const float* __restrict__ tx,
                      const float* __restrict__ tau_adp,
                      const float* __restrict__ tau_m,
                      float* __restrict__ ty,
                      int T, int BN, int N)
{
    __shared__ float fifo[STAGES][BLOCK];

    const int tid = threadIdx.x;
    const int e   = blockIdx.x * BLOCK + tid;   // flattened (b, n)
    if (e >= BN) return;
    const int n = e % N;

    // Per-neuron decay constants (once per thread; negligible).
    const float ro       = expf(-1.0f / tau_adp[n]);
    const float alpha    = expf(-1.0f / tau_m[n]);
    const float one_m_ro = 1.0f - ro;

    // LDS byte address of this lane's FIFO column: flat LDS pointers carry
    // the DS offset in the low 32 bits (aperture above), so truncation gives
    // the VDST address global_load_async_to_lds expects.
    const unsigned lds0 = (unsigned)(uintptr_t)&fifo[0][tid];

    // ---- Prologue: fill the FIFO with timesteps 0..STAGES-1 ----
    const float* gload = tx + e;
#pragma unroll
    for (int s = 0; s < STAGES; ++s) {
        if (s < T) {
            unsigned la = lds0 + (unsigned)(s * BLOCK * 4);
            asm volatile("global_load_async_to_lds_b32 %0, %1, off"
                         :: "v"(la), "v"(gload) : "memory");
        }
        gload += BN;
    }

    // ---- Recurrence state ----
    float v = 0.0f;    // REST
    float y = 0.0f;
    float b = 0.01f;   // B0
    float* out = ty + e;

    auto step = [&](float x) {
        // Same op order as reference (f32):
        //   b = ro*b + (1-ro)*y;  Bth = 0.3 + 1.8*b
        //   v = v*alpha + x - Bth*y;  y = (v > Bth)
        b = ro * b + one_m_ro * y;
        float Bth = 0.3f + 1.8f * b;
        v = v * alpha + x - Bth * y;
        y = (v > Bth) ? 1.0f : 0.0f;
        __builtin_nontemporal_store(y, out);   // streaming write, skip L2 churn
        out += BN;
    };

    // ---- Main loop: 4 timesteps per group, reissue 4 async loads ----
    int t = 0;
    const int tMain = T - STAGES;              // 240 here
    for (; t < tMain; t += 4) {
        // In-order completion: <= STAGES-4 outstanding => loads t..t+3 landed.
        asm volatile("s_wait_asynccnt %0" :: "n"(STAGES - 4) : "memory");

        const int s0 = t & (STAGES - 1);       // 0,4,8,12 -> no wrap in group
        float x0 = fifo[s0 + 0][tid];
        float x1 = fifo[s0 + 1][tid];
        float x2 = fifo[s0 + 2][tid];
        float x3 = fifo[s0 + 3][tid];

        // Slots consumed into VGPRs; safe to let the async engine overwrite.
        asm volatile("s_wait_dscnt 0" ::: "memory");

        unsigned la = lds0 + (unsigned)(s0 * BLOCK * 4);
        asm volatile("global_load_async_to_lds_b32 %0, %1, off"
                     :: "v"(la),                       "v"(gload)          : "memory");
        asm volatile("global_load_async_to_lds_b32 %0, %1, off"
                     :: "v"(la + 1u * BLOCK * 4),      "v"(gload + 1 * BN) : "memory");
        asm volatile("global_load_async_to_lds_b32 %0, %1, off"
                     :: "v"(la + 2u * BLOCK * 4),      "v"(gload + 2 * (size_t)BN) : "memory");
        asm volatile("global_load_async_to_lds_b32 %0, %1, off"
                     :: "v"(la + 3u * BLOCK * 4),      "v"(gload + 3 * (size_t)BN) : "memory");
        gload += 4 * (size_t)BN;

        step(x0); step(x1); step(x2); step(x3);
    }

    // ---- Drain tail: last STAGES steps, no reissues ----
    asm volatile("s_wait_asynccnt 0" ::: "memory");
    for (; t < T; ++t) {
        float x = fifo[t & (STAGES - 1)][tid];
        step(x);
    }
}

extern "C" void kernel_launch(void* const* d_in, const int* in_sizes, int n_in,
                              void* d_out, int out_size, void* d_ws, size_t ws_size,
                              hipStream_t stream) {
    (void)n_in; (void)out_size; (void)d_ws; (void)ws_size;

    const float* tx      = (const float*)d_in[0];
    const float* tau_adp = (const float*)d_in[1];
    const float* tau_m   = (const float*)d_in[2];
    float*       ty      = (float*)d_out;

    const int N  = in_sizes[1];        // 512
    const int T  = 256;                // fixed by reference setup
    const int BN = in_sizes[0] / T;    // B*N = 65536

    dim3 grid((BN + BLOCK - 1) / BLOCK);
    alif_scan_kernel<<<grid, dim3(BLOCK), 0, stream>>>(tx, tau_adp, tau_m, ty,
                                                       T, BN, N);
}